// ClementsBell_Arct_56358560858271
// MI455X (gfx1250) — compile-verified
//
#include <hip/hip_runtime.h>

typedef float v2f __attribute__((ext_vector_type(2)));
typedef float v8f __attribute__((ext_vector_type(8)));

#define NN 1024
#define STEPS 512
#define COLS 16
#define RSTR 17      // LDS row stride (16 cols + 1 pad)
#define LROWS 1025   // rows 0..1023 + pad row 1024 for the shifted (odd) layer

__global__ __launch_bounds__(256) void clements_evolve(const float* __restrict__ ph,
                                                       float* __restrict__ out) {
  __shared__ float  Sre[LROWS * RSTR];
  __shared__ float  Sim[LROWS * RSTR];
  __shared__ float4 Tab[2 * 512];   // per-step (u,v) per pair: [layer][pair] = (ur,ui,vr,vi)

  const int t    = threadIdx.x;
  const int w    = blockIdx.x;      // column tile: global cols [16w, 16w+16)
  const int lane = t & 31;
  const int wave = t >> 5;          // 0..7
  const int r    = lane & 15;       // row-in-block (also N index for B/C operands)
  const int hi   = lane >> 4;       // lane half
  const int rm   = r & 3;
  const bool active = (hi == (rm >> 1));      // this lane holds A elements for row r
  const int  jd     = rm & 1;                 // component index of the diagonal element
  const float sgn   = (r & 1) ? -1.0f : 1.0f; // +u on pair-top row, -u on pair-bottom
  const int  myk    = r >> 2;                 // which K-chunk holds this lane's A entries

  // ---- init: zero LDS, then M0 = diag(e^{i*phases[0]}) restricted to our columns ----
  for (int i = t; i < LROWS * RSTR; i += 256) { Sre[i] = 0.0f; Sim[i] = 0.0f; }
  __syncthreads();
  if (t < COLS) {
    int c = w * COLS + t;
    float sp, cp; __sincosf(ph[c], &sp, &cp);
    Sre[c * RSTR + t] = cp;
    Sim[c * RSTR + t] = sp;
  }
  __syncthreads();

  for (int s = 0; s < STEPS; ++s) {
    // ---- cooperative (u,v) table for both layers of this step ----
    // even layer pair p: rows (2p, 2p+1) with phases pa = phases[1+2s]
    // odd  layer pair p: rows (2p+1, 2p+2) with phases pb = phases[2+2s]
    //   p=511 is the boundary pseudo-pair (row 1023, pad): u = e^{i*pb[1023]}, v = 0
    for (int idx = t; idx < 1024; idx += 256) {
      const int layer = idx >> 9;
      const int p     = idx & 511;
      const float* prow = ph + (size_t)(1 + 2 * s + layer) * NN;
      float ur, ui, vr, vi;
      if (layer == 1 && p == 511) {
        float sa, ca; __sincosf(prow[NN - 1], &sa, &ca);
        ur = ca; ui = sa; vr = 0.0f; vi = 0.0f;
      } else {
        const int i0 = 2 * p + layer;
        float sa, ca, sb, cb;
        __sincosf(prow[i0],     &sa, &ca);
        __sincosf(prow[i0 + 1], &sb, &cb);
        ur = 0.5f * (ca - cb);  ui = 0.5f * (sa - sb);   // u = (alpha-beta)/2
        vr = -0.5f * (sa + sb); vi = 0.5f * (ca + cb);   // v = i(alpha+beta)/2
      }
      Tab[idx] = make_float4(ur, ui, vr, vi);
    }
    __syncthreads();

    #pragma unroll
    for (int layer = 0; layer < 2; ++layer) {
      const int base = layer;  // 0: even pairs, 1: odd pairs (row grid shifted by 1)

      // Odd layer: row 0 is identity in O, so it only gets the heater phase e^{i*pb[0]}.
      if (layer == 1 && t < COLS) {
        float s0, c0; __sincosf(ph[(size_t)(2 + 2 * s) * NN], &s0, &c0);
        float xr = Sre[t], xi = Sim[t];
        Sre[t] = c0 * xr - s0 * xi;
        Sim[t] = s0 * xr + c0 * xi;
      }

      for (int bb = 0; bb < 8; ++bb) {
        const int b     = wave + 8 * bb;   // row-block 0..63 (disjoint across waves)
        const int rbase = base + 16 * b;   // first global row of this block

        // ---- fetch this lane's pair coefficients, place in A-operand layout ----
        float dre = 0.0f, dim_ = 0.0f, ore = 0.0f, oim = 0.0f;
        if (active) {
          const float4 uv = Tab[layer * 512 + 8 * b + (r >> 1)];
          dre = sgn * uv.x; dim_ = sgn * uv.y; ore = uv.z; oim = uv.w;
        }
        const float c0re = jd ? ore  : dre,  c1re = jd ? dre  : ore;
        const float c0im = jd ? oim  : dim_, c1im = jd ? dim_ : oim;

        v2f Ar[4], Ai[4];
        #pragma unroll
        for (int k = 0; k < 4; ++k) {
          const bool m = active && (myk == k);
          Ar[k][0] = m ? c0re : 0.0f;  Ar[k][1] = m ? c1re : 0.0f;
          Ai[k][0] = m ? c0im : 0.0f;  Ai[k][1] = m ? c1im : 0.0f;
        }

        // ---- load X in B-operand layout: VGPR j <-> K = j + 2*hi within each chunk ----
        const int col = r;
        v2f Xr[4], Xi[4];
        #pragma unroll
        for (int k = 0; k < 4; ++k) {
          const int row0 = rbase + 4 * k + 2 * hi;
          Xr[k][0] = Sre[row0 * RSTR + col];
          Xr[k][1] = Sre[(row0 + 1) * RSTR + col];
          Xi[k][0] = Sim[row0 * RSTR + col];
          Xi[k][1] = Sim[(row0 + 1) * RSTR + col];
        }

        // ---- complex D = A@X via 16 chained V_WMMA_F32_16X16X4_F32 ----
        v8f Cim = {0,0,0,0,0,0,0,0};
        #pragma unroll
        for (int k = 0; k < 4; ++k)
          Cim = __builtin_amdgcn_wmma_f32_16x16x4_f32(false, Ar[k], false, Xi[k],
                                                      (short)0, Cim, false, false);
        #pragma unroll
        for (int k = 0; k < 4; ++k)
          Cim = __builtin_amdgcn_wmma_f32_16x16x4_f32(false, Ai[k], false, Xr[k],
                                                      (short)0, Cim, false, false);
        v8f Cre = {0,0,0,0,0,0,0,0};
        #pragma unroll
        for (int k = 0; k < 4; ++k)
          Cre = __builtin_amdgcn_wmma_f32_16x16x4_f32(false, Ar[k], false, Xr[k],
                                                      (short)0, Cre, false, false);
        #pragma unroll
        for (int k = 0; k < 4; ++k) { Ai[k][0] = -Ai[k][0]; Ai[k][1] = -Ai[k][1]; }
        #pragma unroll
        for (int k = 0; k < 4; ++k)
          Cre = __builtin_amdgcn_wmma_f32_16x16x4_f32(false, Ai[k], false, Xi[k],
                                                      (short)0, Cre, false, false);

        // ---- store D back (C-layout: VGPR j -> row j (lanes 0-15) / j+8 (lanes 16-31)) ----
        #pragma unroll
        for (int j = 0; j < 8; ++j) {
          const int row = rbase + j + 8 * hi;
          Sre[row * RSTR + col] = Cre[j];
          Sim[row * RSTR + col] = Cim[j];
        }
      }
      __syncthreads();
    }
  }

  // ---- final heater (phases[N+1]) + writeout as interleaved complex64 ----
  const int cl  = t & 15;
  const int rr0 = t >> 4;
  for (int i = 0; i < 64; ++i) {
    const int rrow = rr0 + 16 * i;
    float sf, cf; __sincosf(ph[(size_t)(NN + 1) * NN + rrow], &sf, &cf);
    const float xr = Sre[rrow * RSTR + cl];
    const float xi = Sim[rrow * RSTR + cl];
    const int c = w * COLS + cl;
    float2 val;
    val.x = cf * xr - sf * xi;
    val.y = sf * xr + cf * xi;
    *(float2*)(out + ((size_t)rrow * NN + c) * 2) = val;
  }
}

extern "C" void kernel_launch(void* const* d_in, const int* in_sizes, int n_in,
                              void* d_out, int out_size, void* d_ws, size_t ws_size,
                              hipStream_t stream) {
  (void)in_sizes; (void)n_in; (void)d_ws; (void)ws_size; (void)out_size;
  const float* phases = (const float*)d_in[0];     // (1026, 1024) float32
  float* out = (float*)d_out;                      // 1024x1024 complex64, interleaved
  clements_evolve<<<dim3(64), dim3(256), 0, stream>>>(phases, out);
}